// EaPool_64828236366591
// MI455X (gfx1250) — compile-verified
//
#include <hip/hip_runtime.h>
#include <cstdint>

// EaPool: per-2x2-window select of max-pool vs avg-pool, selector from gene.
// x: [16, 64, 256, 256] f32   gene: [128*128] i32   out: [16, 64, 128, 128] f32
//
// Memory-bound (~335 MB total -> ~14.4 us at 23.3 TB/s). WMMA is inapplicable
// (max is non-linear); CDNA5 leverage = async global->LDS data mover for the
// input stream, NT cache hints on the single-use streams (input + output),
// regular cached load for the L2-resident gene table.

#define B_     16
#define C_     64
#define H_     256
#define W_     256
#define OH_    (H_ / 2)   // 128
#define OW_    (W_ / 2)   // 128
#define PLANE_IN  (H_ * W_)     // 65536 floats
#define PLANE_OUT (OH_ * OW_)   // 16384 floats

typedef float v2f __attribute__((ext_vector_type(2)));

__global__ __launch_bounds__(256) void eapool_kernel(
    const float* __restrict__ x,
    const int*   __restrict__ gene,
    float*       __restrict__ out)
{
    // Two float4 row-segments per thread, interleaved to keep ds_load_b128
    // bank-conflict free: row0 of all threads first, then row1.
    __shared__ float4 tile[512];   // 8 KB

    const int tid   = threadIdx.x;
    const int bid   = blockIdx.x;
    const int plane = bid >> 5;    // [0, 1024): b*C + c
    const int rg    = bid & 31;    // row-group: 4 output rows each

    const int ri   = tid >> 6;     // output row within group [0,4)
    const int j2   = tid & 63;     // window-pair column [0,64)
    const int orow = rg * 4 + ri;  // output row [0,128)
    const int irow = orow * 2;     // input row

    const float* src0 = x + (size_t)plane * PLANE_IN + (size_t)irow * W_ + j2 * 4;
    const float* src1 = src0 + W_;

    // LDS byte addresses (generic shared pointer low 32 bits == LDS offset).
    unsigned lds0 = (unsigned)(uintptr_t)(&tile[tid]);
    unsigned lds1 = (unsigned)(uintptr_t)(&tile[tid + 256]);

    // Async DMA of both 16B row segments into LDS (ASYNCcnt path).
    // Input is read exactly once and exceeds L2 -> non-temporal.
    asm volatile(
        "global_load_async_to_lds_b128 %0, %2, off th:TH_LOAD_NT\n\t"
        "global_load_async_to_lds_b128 %1, %3, off th:TH_LOAD_NT"
        :
        : "v"(lds0), "v"(lds1), "v"(src0), "v"(src1)
        : "memory");

    // Overlap: fetch the two gene selectors (L2-resident, reused by all planes).
    const int2 g = *(const int2*)(gene + orow * OW_ + j2 * 2);

    asm volatile("s_wait_asynccnt 0" ::: "memory");

    // Same-wave consumption of async-written LDS: no barrier needed after wait.
    const float4 r0 = tile[tid];         // input row irow,   4 floats
    const float4 r1 = tile[tid + 256];   // input row irow+1, 4 floats

    // Window 0: {r0.x, r0.y, r1.x, r1.y}; Window 1: {r0.z, r0.w, r1.z, r1.w}
    const float m0 = fmaxf(fmaxf(r0.x, r0.y), fmaxf(r1.x, r1.y));
    const float a0 = (r0.x + r0.y + r1.x + r1.y) * 0.25f;
    const float m1 = fmaxf(fmaxf(r0.z, r0.w), fmaxf(r1.z, r1.w));
    const float a1 = (r0.z + r0.w + r1.z + r1.w) * 0.25f;

    v2f o;
    o.x = (g.x == 0) ? m0 : a0;
    o.y = (g.y == 0) ? m1 : a1;

    // Write-once output: non-temporal store, don't pollute L2.
    v2f* dst = (v2f*)(out + (size_t)plane * PLANE_OUT + (size_t)orow * OW_ + j2 * 2);
    __builtin_nontemporal_store(o, dst);
}

extern "C" void kernel_launch(void* const* d_in, const int* in_sizes, int n_in,
                              void* d_out, int out_size, void* d_ws, size_t ws_size,
                              hipStream_t stream) {
    const float* x    = (const float*)d_in[0];
    const int*   gene = (const int*)d_in[1];
    float*       out  = (float*)d_out;

    // 1024 planes * 32 row-groups; 256 threads, 2 windows/thread.
    dim3 grid(B_ * C_ * (OH_ / 4));
    dim3 block(256);
    hipLaunchKernelGGL(eapool_kernel, grid, block, 0, stream, x, gene, out);
}